// Perceptron_38130719654349
// MI455X (gfx1250) — compile-verified
//
#include <hip/hip_runtime.h>

// ---------------------------------------------------------------------------
// out[i] = W[0, idx[i]] + W[1, idx[i]] + W[2, idx[i]] + (b0+b1+b2)
//
// Pure gather: memory-bound (~0.01 FLOP/byte -> WMMA is provably useless;
// HBM floor ~11.5 us at 23.3 TB/s). Strategy:
//   Pass 1: fold W's 3 rows + bias into a 400 KB LUT S[v] in d_ws (L2-resident).
//   Pass 2: 128-bit NT streaming gather: out[i] = S[idx[i]], 8 elems/thread.
// Non-temporal hints keep the single-touch index/output streams from evicting
// the hot LUT; LUT reads stay regular-temporal so they hit L2/WGP$.
// All index math is 32-bit (n = 33.5M), and the fast/tail split is a uniform
// per-block scalar branch so the hot path has zero divergence overhead.
// ---------------------------------------------------------------------------

typedef int   vint4  __attribute__((ext_vector_type(4)));
typedef float vflt4  __attribute__((ext_vector_type(4)));

// Pass 2 (listed first so its disasm shows in compile feedback):
// 8 elements/thread: 2x NT global_load_b128 (indices), 8x RT global_load_b32
// (LUT, L2-hot), 2x NT global_store_b128 (output), + global_prefetch_b8.
__global__ void gather_lut_kernel(const int* __restrict__ idx,
                                  const float* __restrict__ S,
                                  float* __restrict__ out,
                                  unsigned n) {
    const unsigned t    = blockIdx.x * blockDim.x + threadIdx.x;
    const unsigned base = t * 8u;

    // Uniform (scalar) condition: is this block fully covered?
    const unsigned blockEnd = (blockIdx.x + 1u) * (blockDim.x * 8u);

    if (blockEnd <= n) {
        // Speculative prefetch of the index stream (OOB silently dropped).
        __builtin_prefetch(idx + base + 16384u, 0, 1);

        vint4 a = __builtin_nontemporal_load((const vint4*)(idx + base));
        vint4 c = __builtin_nontemporal_load((const vint4*)(idx + base + 4u));

        vflt4 o0, o1;
        o0.x = S[a.x];
        o0.y = S[a.y];
        o0.z = S[a.z];
        o0.w = S[a.w];
        o1.x = S[c.x];
        o1.y = S[c.y];
        o1.z = S[c.z];
        o1.w = S[c.w];

        __builtin_nontemporal_store(o0, (vflt4*)(out + base));
        __builtin_nontemporal_store(o1, (vflt4*)(out + base + 4u));
    } else {
        // Ragged last block (n is 8-divisible for the reference; kept for safety).
        const unsigned end = (base + 8u < n) ? (base + 8u) : n;
#pragma clang loop unroll(disable) vectorize(disable)
        for (unsigned i = base; i < end; ++i) {
            out[i] = S[idx[i]];
        }
    }
}

// Pass 1: S[v] = W[v] + W[V+v] + W[2V+v] + (b0+b1+b2)
__global__ void lut_build_kernel(const float* __restrict__ W,
                                 const float* __restrict__ b,
                                 float* __restrict__ S,
                                 int V) {
    int v = blockIdx.x * blockDim.x + threadIdx.x;
    if (v < V) {
        float bsum = b[0] + b[1] + b[2];
        S[v] = W[v] + W[V + v] + W[2 * V + v] + bsum;
    }
}

extern "C" void kernel_launch(void* const* d_in, const int* in_sizes, int n_in,
                              void* d_out, int out_size, void* d_ws, size_t ws_size,
                              hipStream_t stream) {
    // setup_inputs order: input (int32 idx, 16384*2048), W (3*V f32), b (3 f32)
    const int*   idx = (const int*)d_in[0];
    const float* W   = (const float*)d_in[1];
    const float* b   = (const float*)d_in[2];
    float*       out = (float*)d_out;
    float*       S   = (float*)d_ws;              // V * 4 bytes = 400 KB

    const int      V = in_sizes[1] / 3;           // 100000
    const unsigned n = (unsigned)out_size;        // 33,554,432

    // Pass 1: build LUT (391 blocks x 256 threads).
    lut_build_kernel<<<(V + 255) / 256, 256, 0, stream>>>(W, b, S, V);

    // Pass 2: streaming gather. 8 elems/thread, 256 threads/block (8 wave32s).
    const unsigned n8     = (n + 7u) >> 3;
    const unsigned blocks = (n8 + 255u) / 256u;
    gather_lut_kernel<<<blocks, 256, 0, stream>>>(idx, S, out, n);
}